// DistGraphConv_53257594470854
// MI455X (gfx1250) — compile-verified
//
#include <hip/hip_runtime.h>
#include <hip/hip_bf16.h>

// Problem constants (match reference)
#define PPART 4
#define NIN   100000
#define MOUT  50000
#define EDG   400000
#define DD    128
#define MTILES (MOUT / 16)   // 3125

typedef __attribute__((ext_vector_type(2))) float v2f;
typedef __attribute__((ext_vector_type(8))) float v8f;

// ---------------------------------------------------------------------------
// Kernel 1: edge scatter. One wave32 per edge: coalesced 512B row gather,
// float atomics into neigh_sum[dst], lane 0 counts degree.
// ---------------------------------------------------------------------------
__global__ void dgc_edge_scatter(const float* __restrict__ x,     // [NIN, DD]
                                 const int*   __restrict__ esrc,  // [EDG]
                                 const int*   __restrict__ edst,  // [EDG]
                                 float* __restrict__ nsum,        // [MOUT, DD]
                                 float* __restrict__ deg)         // [MOUT]
{
    int gtid = blockIdx.x * blockDim.x + threadIdx.x;
    int wave = gtid >> 5;
    int lane = gtid & 31;
    if (wave >= EDG) return;

    int s = esrc[wave];
    int d = edst[wave];

    const float4* xrow = (const float4*)(x + (size_t)s * DD);
    float4 v = xrow[lane];                       // cols 4*lane .. 4*lane+3

    float* nrow = nsum + (size_t)d * DD + lane * 4;
    atomicAdd(nrow + 0, v.x);
    atomicAdd(nrow + 1, v.y);
    atomicAdd(nrow + 2, v.z);
    atomicAdd(nrow + 3, v.w);
    if (lane == 0) atomicAdd(deg + d, 1.0f);
}

// ---------------------------------------------------------------------------
// Kernel 2: fused SAGE GEMM + bias + deg-normalize + merge scatter.
//   out[orow] += x[:M] @ Wself + (nsum/max(deg,1)) @ Wneigh + bias
//
// 2 waves/block (64 threads), each wave owns a 16-column strip. The wave's
// two 128x16 weight panels are staged in LDS in WMMA B-fragment layout
// (fixes the round-1 scratch spill): lane L's (k-step, matrix) v2f sits at
// dword lane*2 of a 64-dword row -> ds_load_b64 touches all 64 banks once.
// Each wave reads only the region it wrote, so no workgroup barrier needed.
//
// Fragment layouts (ISA 7.12.2, 32-bit):
//   A 16x4 : lane holds row (lane&15); v0=K[kb], v1=K[kb+1], kb=2*(lane>>4)
//   B 4x16 : lane holds col (lane&15); v0=K[kb],  v1=K[kb+1]
//   C 16x16: VGPR i -> row 8*(lane>>4)+i, col (lane&15)
// ---------------------------------------------------------------------------
__global__ __launch_bounds__(64)
void dgc_sage_gemm(const float* __restrict__ xs,     // [NIN, DD] (rows 0..M-1 for self)
                   const float* __restrict__ nsum,   // [MOUT, DD]
                   const float* __restrict__ deg,    // [MOUT]
                   const float* __restrict__ Wself,  // [DD, DD]
                   const float* __restrict__ Wneigh, // [DD, DD]
                   const float* __restrict__ bias,   // [DD]
                   const int*   __restrict__ midx,   // [MOUT] merge indices
                   float* __restrict__ out,          // [MOUT, DD] dest partition
                   int isDiag)
{
    // [wave][matrix][kstep][lane*2 + {0,1}] : 2*2*32*64*4B = 32 KB
    __shared__ float ldsB[2][2][32][64];

    const int lane = threadIdx.x & 31;
    const int wave = threadIdx.x >> 5;
    const int col  = blockIdx.x * 32 + wave * 16 + (lane & 15); // global N column
    const int kb   = (lane >> 4) << 1;                          // 0 or 2

    // Stage this wave's B fragments into LDS (read once from global).
    float* myS = &ldsB[wave][0][0][lane * 2];
    float* myN = &ldsB[wave][1][0][lane * 2];
#pragma unroll
    for (int i = 0; i < 32; ++i) {
        int k = i * 4 + kb;
        myS[i * 64 + 0] = Wself [(size_t)(k    ) * DD + col];
        myS[i * 64 + 1] = Wself [(size_t)(k + 1) * DD + col];
        myN[i * 64 + 0] = Wneigh[(size_t)(k    ) * DD + col];
        myN[i * 64 + 1] = Wneigh[(size_t)(k + 1) * DD + col];
    }

    const float bcol = bias[col];
    const int   rhi  = (lane >> 4) << 3;

    for (int tile = blockIdx.y; tile < MTILES; tile += gridDim.y) {
        const int rowbase = tile * 16;
        const int arow    = rowbase + (lane & 15);
        const float rdeg  = 1.0f / fmaxf(deg[arow], 1.0f);
        const float* xrow = xs   + (size_t)arow * DD;
        const float* nrow = nsum + (size_t)arow * DD;

        // Prefetch next tile's activation rows while this tile computes.
        const int ntile = tile + gridDim.y;
        if (ntile < MTILES) {
            const size_t nr = (size_t)(ntile * 16 + (lane & 15)) * DD;
            const int poff  = (lane >> 4) * 64;       // bytes 0/256 per row half
            __builtin_prefetch(xs   + nr + poff,      0, 0);
            __builtin_prefetch(xs   + nr + poff + 32, 0, 0);
            __builtin_prefetch(nsum + nr + poff,      0, 0);
            __builtin_prefetch(nsum + nr + poff + 32, 0, 0);
        }

        v8f c = {};
#pragma unroll
        for (int i = 0; i < 32; ++i) {
            int k = i * 4 + kb;
            v2f aS = *(const v2f*)(xrow + k);                 // global_load_b64
            v2f aNraw = *(const v2f*)(nrow + k);
            v2f aN; aN.x = aNraw.x * rdeg; aN.y = aNraw.y * rdeg;
            v2f bS = *(const v2f*)(myS + i * 64);             // ds_load_b64
            v2f bN = *(const v2f*)(myN + i * 64);
            c = __builtin_amdgcn_wmma_f32_16x16x4_f32(
                    false, aS, false, bS, (short)0, c, false, false);
            c = __builtin_amdgcn_wmma_f32_16x16x4_f32(
                    false, aN, false, bN, (short)0, c, false, false);
        }

        // Fused merge: diagonal at identity rows, off-diagonal via
        // merge_indices; out is zero-initialized so everything accumulates.
#pragma unroll
        for (int i = 0; i < 8; ++i) {
            int m    = rowbase + rhi + i;
            int orow = isDiag ? m : midx[m];
            atomicAdd(out + (size_t)orow * DD + col, c[i] + bcol);
        }
    }
}

// ---------------------------------------------------------------------------
// Host launcher: 16 (s,d) pairs; per pair zero the neigh_sum/deg workspace,
// scatter edges, then fused GEMM+merge. Output zeroed once up front.
// ---------------------------------------------------------------------------
extern "C" void kernel_launch(void* const* d_in, const int* in_sizes, int n_in,
                              void* d_out, int out_size, void* d_ws, size_t ws_size,
                              hipStream_t stream)
{
    const float* x    = (const float*)d_in[0];   // [P, N, D]
    const float* Wsl  = (const float*)d_in[1];   // [P, D, D]
    const float* Wng  = (const float*)d_in[2];   // [P, D, D]
    const float* bv   = (const float*)d_in[3];   // [P, D]
    const int*   esrc = (const int*)  d_in[4];   // [P, P, E]
    const int*   edst = (const int*)  d_in[5];   // [P, P, E]
    const int*   midx = (const int*)  d_in[6];   // [P, P, M]
    float*       out  = (float*)d_out;           // [P, M, D]

    float* nsum = (float*)d_ws;                  // [M, D]
    float* deg  = nsum + (size_t)MOUT * DD;      // [M]

    hipMemsetAsync(out, 0, (size_t)PPART * MOUT * DD * sizeof(float), stream);

    const int scatterBlocks = (EDG * 32 + 255) / 256;   // one wave per edge
    const dim3 gemmGrid(DD / 32, 125);                  // 4 col groups x 125 (25 tiles/block)

    for (int s = 0; s < PPART; ++s) {
        const float* xs = x + (size_t)s * NIN * DD;
        for (int d = 0; d < PPART; ++d) {
            hipMemsetAsync(nsum, 0,
                           ((size_t)MOUT * DD + MOUT) * sizeof(float), stream);

            const size_t epair = ((size_t)s * PPART + d) * EDG;
            dgc_edge_scatter<<<scatterBlocks, 256, 0, stream>>>(
                xs, esrc + epair, edst + epair, nsum, deg);

            dgc_sage_gemm<<<gemmGrid, 64, 0, stream>>>(
                xs, nsum, deg,
                Wsl + (size_t)s * DD * DD,
                Wng + (size_t)s * DD * DD,
                bv  + (size_t)s * DD,
                midx + ((size_t)s * PPART + d) * MOUT,
                out + (size_t)d * MOUT * DD,
                (s == d) ? 1 : 0);
        }
    }
}